// SlotAttention_23407571764042
// MI455X (gfx1250) — compile-verified
//
#include <hip/hip_runtime.h>
#include <hip/hip_bf16.h>

typedef __attribute__((ext_vector_type(2))) float v2f;
typedef __attribute__((ext_vector_type(8))) float v8f;

#define BB 32
#define FF 4096
#define SS 8
#define DD 512
#define HH 2048
#define NITER 3
#define EPS_A 1e-8f
#define LN_EPS_C 1e-5f
#define SCALE_C 0.04419417382415922f   // 512^-0.5

// ---------------------------------------------------------------- WMMA core
__device__ __forceinline__ v8f wmma4(v2f a, v2f b, v8f c) {
  // V_WMMA_F32_16X16X4_F32 : D = A(16x4) * B(4x16) + C(16x16), fp32 exact
  return __builtin_amdgcn_wmma_f32_16x16x4_f32(
      /*neg_a=*/false, a, /*neg_b=*/false, b,
      /*c_mod=*/(short)0, c, /*reuse_a=*/false, /*reuse_b=*/false);
}

// A tile in LDS: [16 rows][lda] row-major.  lanes0-15 -> K0/K1, lanes16-31 ->
// K2/K3, M = lane&15.  kk is always even -> one 8B-aligned ds_load_b64.
__device__ __forceinline__ v2f frag_a(const float* A, int lda, int lane, int k0) {
  int row = lane & 15;
  int kk  = k0 + ((lane >> 4) << 1);
  return *(const v2f*)&A[row * lda + kk];
}
// B tile in LDS, k-pair interleaved: element (k,n) at [((k>>1)*ldb + n)*2 + (k&1)].
// Fragment = contiguous {k, k+1} pair -> one ds_load_b64, no register packing.
__device__ __forceinline__ v2f frag_b(const float* B, int ldb, int lane, int k0, int n0) {
  int col = n0 + (lane & 15);
  int p   = (k0 >> 1) + (lane >> 4);
  return *(const v2f*)&B[(p * ldb + col) * 2];
}

__device__ __forceinline__ float wave_sum(float v) {
#pragma unroll
  for (int o = 16; o; o >>= 1) v += __shfl_xor(v, o, 32);
  return v;
}

// ------------------------------------------------- per-row LayerNorm stats
__global__ void __launch_bounds__(256)
row_stats_kernel(const float* __restrict__ X, float* __restrict__ mean,
                 float* __restrict__ rstd) {
  int wv = threadIdx.x >> 5, lane = threadIdx.x & 31;
  int row = blockIdx.x * 8 + wv;
  const float* x = X + (size_t)row * DD;
  float s = 0.f, s2 = 0.f;
#pragma unroll
  for (int j = 0; j < DD / 32; ++j) {
    float v = x[lane + 32 * j]; s += v; s2 += v * v;
  }
  s = wave_sum(s); s2 = wave_sum(s2);
  if (lane == 0) {
    float m = s * (1.f / DD);
    float var = s2 * (1.f / DD) - m * m;
    mean[row] = m;
    rstd[row] = rsqrtf(var + LN_EPS_C);
  }
}

// ------------------------------------- LayerNorm of small [256,512] tensors
__global__ void __launch_bounds__(256)
ln_rows_kernel(const float* __restrict__ X, const float* __restrict__ w,
               const float* __restrict__ b, float* __restrict__ Y) {
  int wv = threadIdx.x >> 5, lane = threadIdx.x & 31;
  int row = blockIdx.x * 8 + wv;
  const float* x = X + (size_t)row * DD;
  float xv[16]; float s = 0.f, s2 = 0.f;
#pragma unroll
  for (int j = 0; j < 16; ++j) {
    float v = x[lane + 32 * j]; xv[j] = v; s += v; s2 += v * v;
  }
  s = wave_sum(s); s2 = wave_sum(s2);
  float m = s * (1.f / DD);
  float r = rsqrtf(s2 * (1.f / DD) - m * m + LN_EPS_C);
  float* y = Y + (size_t)row * DD;
#pragma unroll
  for (int j = 0; j < 16; ++j) {
    int d = lane + 32 * j;
    y[d] = (xv[j] - m) * r * w[d] + b[d];
  }
}

// ---------------- fused LN + dual GEMM:  keys = LN(X)@Wk, values = LN(X)@Wv
// X: [B*F, 512]; tile 128(M) x 64(N); 8 waves each 16x64; K chunk 16.
__global__ void __launch_bounds__(256)
gemm_ln_dual_kernel(const float* __restrict__ X, const float* __restrict__ mean,
                    const float* __restrict__ rstd, const float* __restrict__ lnw,
                    const float* __restrict__ lnb, const float* __restrict__ Wk,
                    const float* __restrict__ Wv, float* __restrict__ Ko,
                    float* __restrict__ Vo) {
  __shared__ float As[128 * 16];
  __shared__ float Bks[16 * 64];
  __shared__ float Bvs[16 * 64];
  int t = threadIdx.x, lane = t & 31, wv = t >> 5;
  int m0 = blockIdx.x * 128, n0 = blockIdx.y * 64;

  int arow = t >> 1, aoff = (t & 1) * 8;
  size_t grow = (size_t)(m0 + arow);
  float mn = mean[grow], rs = rstd[grow];
  const float* xrow = X + grow * DD;
  int bp = t >> 5, bn2 = (t & 31) * 2;   // k-pair, column pair

  v8f accK[4] = {}, accV[4] = {};
  for (int kc = 0; kc < DD; kc += 16) {
    __syncthreads();
#pragma unroll
    for (int j = 0; j < 8; ++j) {
      int k = kc + aoff + j;
      As[arow * 16 + aoff + j] = (xrow[k] - mn) * rs * lnw[k] + lnb[k];
    }
    {
      float2 r0 = *(const float2*)&Wk[(size_t)(kc + 2 * bp) * DD + n0 + bn2];
      float2 r1 = *(const float2*)&Wk[(size_t)(kc + 2 * bp + 1) * DD + n0 + bn2];
      float4 wk; wk.x = r0.x; wk.y = r1.x; wk.z = r0.y; wk.w = r1.y;
      *(float4*)&Bks[(bp * 64 + bn2) * 2] = wk;
      float2 s0 = *(const float2*)&Wv[(size_t)(kc + 2 * bp) * DD + n0 + bn2];
      float2 s1 = *(const float2*)&Wv[(size_t)(kc + 2 * bp + 1) * DD + n0 + bn2];
      float4 wvv; wvv.x = s0.x; wvv.y = s1.x; wvv.z = s0.y; wvv.w = s1.y;
      *(float4*)&Bvs[(bp * 64 + bn2) * 2] = wvv;
    }
    __syncthreads();
    const float* Aw = &As[(wv * 16) * 16];
#pragma unroll
    for (int ks = 0; ks < 16; ks += 4) {
      v2f a = frag_a(Aw, 16, lane, ks);
#pragma unroll
      for (int nt = 0; nt < 4; ++nt) {
        v2f bk = frag_b(Bks, 64, lane, ks, nt * 16);
        accK[nt] = wmma4(a, bk, accK[nt]);
        v2f bv = frag_b(Bvs, 64, lane, ks, nt * 16);
        accV[nt] = wmma4(a, bv, accV[nt]);
      }
    }
  }
  int half = lane >> 4, l16 = lane & 15;
#pragma unroll
  for (int nt = 0; nt < 4; ++nt)
#pragma unroll
    for (int i = 0; i < 8; ++i) {
      size_t mrow = (size_t)(m0 + wv * 16 + half * 8 + i);
      int ncol = n0 + nt * 16 + l16;
      Ko[mrow * DD + ncol] = accK[nt][i];
      Vo[mrow * DD + ncol] = accV[nt][i];
    }
}

// ------------- generic GEMM C = A[M,K] @ B[K,N] (+bias, +relu, +residual)
// EPI 0: plain   EPI 1: +bias, relu   EPI 2: +bias, += aux (residual)
template <int EPI>
__global__ void __launch_bounds__(256)
gemm_rm_kernel(const float* __restrict__ A, const float* __restrict__ B,
               const float* __restrict__ bias, float* __restrict__ C,
               const float* __restrict__ aux, int K, int N) {
  __shared__ float As[128 * 16];
  __shared__ float Bs[16 * 64];
  int t = threadIdx.x, lane = t & 31, wv = t >> 5;
  int m0 = blockIdx.x * 128, n0 = blockIdx.y * 64;
  int arow = t >> 1, aoff = (t & 1) * 8;
  const float* xrow = A + (size_t)(m0 + arow) * K;
  int bp = t >> 5, bn2 = (t & 31) * 2;

  v8f acc[4] = {};
  for (int kc = 0; kc < K; kc += 16) {
    __syncthreads();
    {
      float4 a0 = *(const float4*)&xrow[kc + aoff];
      float4 a1 = *(const float4*)&xrow[kc + aoff + 4];
      *(float4*)&As[arow * 16 + aoff] = a0;
      *(float4*)&As[arow * 16 + aoff + 4] = a1;
      float2 r0 = *(const float2*)&B[(size_t)(kc + 2 * bp) * N + n0 + bn2];
      float2 r1 = *(const float2*)&B[(size_t)(kc + 2 * bp + 1) * N + n0 + bn2];
      float4 wb; wb.x = r0.x; wb.y = r1.x; wb.z = r0.y; wb.w = r1.y;
      *(float4*)&Bs[(bp * 64 + bn2) * 2] = wb;
    }
    __syncthreads();
    const float* Aw = &As[(wv * 16) * 16];
#pragma unroll
    for (int ks = 0; ks < 16; ks += 4) {
      v2f a = frag_a(Aw, 16, lane, ks);
#pragma unroll
      for (int nt = 0; nt < 4; ++nt) {
        v2f b = frag_b(Bs, 64, lane, ks, nt * 16);
        acc[nt] = wmma4(a, b, acc[nt]);
      }
    }
  }
  int half = lane >> 4, l16 = lane & 15;
#pragma unroll
  for (int nt = 0; nt < 4; ++nt) {
    int ncol = n0 + nt * 16 + l16;
    float bv = (EPI >= 1) ? bias[ncol] : 0.f;
#pragma unroll
    for (int i = 0; i < 8; ++i) {
      size_t mrow = (size_t)(m0 + wv * 16 + half * 8 + i);
      float v = acc[nt][i] + bv;
      if (EPI == 1) v = fmaxf(v, 0.f);
      if (EPI == 2) v += aux[mrow * (size_t)N + ncol];
      C[mrow * (size_t)N + ncol] = v;
    }
  }
}

// -------- GEMM with transposed weight: C = A[M,K] @ W[N,K]^T + bias[N]
// A float2 along K of one W row is exactly one interleaved k-pair -> direct store.
__global__ void __launch_bounds__(256)
gemm_bt_kernel(const float* __restrict__ A, const float* __restrict__ W,
               const float* __restrict__ bias, float* __restrict__ C,
               int K, int N) {
  __shared__ float As[128 * 16];
  __shared__ float Bs[16 * 64];
  int t = threadIdx.x, lane = t & 31, wv = t >> 5;
  int m0 = blockIdx.x * 128, n0 = blockIdx.y * 64;
  int arow = t >> 1, aoff = (t & 1) * 8;
  const float* xrow = A + (size_t)(m0 + arow) * K;
  int bn = t & 63, bp0 = t >> 6;   // column, k-pair base (0..3)

  v8f acc[4] = {};
  for (int kc = 0; kc < K; kc += 16) {
    __syncthreads();
    {
      float4 a0 = *(const float4*)&xrow[kc + aoff];
      float4 a1 = *(const float4*)&xrow[kc + aoff + 4];
      *(float4*)&As[arow * 16 + aoff] = a0;
      *(float4*)&As[arow * 16 + aoff + 4] = a1;
#pragma unroll
      for (int j = 0; j < 2; ++j) {
        int p = bp0 + j * 4;
        float2 f = *(const float2*)&W[(size_t)(n0 + bn) * K + kc + 2 * p];
        *(float2*)&Bs[(p * 64 + bn) * 2] = f;
      }
    }
    __syncthreads();
    const float* Aw = &As[(wv * 16) * 16];
#pragma unroll
    for (int ks = 0; ks < 16; ks += 4) {
      v2f a = frag_a(Aw, 16, lane, ks);
#pragma unroll
      for (int nt = 0; nt < 4; ++nt) {
        v2f b = frag_b(Bs, 64, lane, ks, nt * 16);
        acc[nt] = wmma4(a, b, acc[nt]);
      }
    }
  }
  int half = lane >> 4, l16 = lane & 15;
#pragma unroll
  for (int nt = 0; nt < 4; ++nt) {
    int ncol = n0 + nt * 16 + l16;
    float bv = bias[ncol];
#pragma unroll
    for (int i = 0; i < 8; ++i) {
      size_t mrow = (size_t)(m0 + wv * 16 + half * 8 + i);
      C[mrow * (size_t)N + ncol] = acc[nt][i] + bv;
    }
  }
}

// -------- dots[b,s,f] = SCALE * q[b,s,:].keys[b,f,:]  (M=8 padded to 16)
__global__ void __launch_bounds__(256)
dots_kernel(const float* __restrict__ Q, const float* __restrict__ Keys,
            float* __restrict__ Dots) {
  __shared__ float As[16 * 16];
  __shared__ float Bs[16 * 512];
  int t = threadIdx.x, lane = t & 31, wv = t >> 5;
  int b = blockIdx.x, nb = blockIdx.y * 512;
  const float* qb = Q + (size_t)b * SS * DD;
  const float* kb = Keys + ((size_t)b * FF + nb) * DD;
  v8f acc[4] = {};
  for (int kc = 0; kc < DD; kc += 16) {
    __syncthreads();
    {
      int r = t >> 4, k = t & 15;
      As[r * 16 + k] = (r < SS) ? qb[(size_t)r * DD + kc + k] : 0.f;
    }
    {
      int k4 = (t & 3) * 4;   // 4 consecutive k = 2 interleaved pairs
#pragma unroll
      for (int j = 0; j < 8; ++j) {
        int n = (t >> 2) + j * 64;
        float4 f = *(const float4*)&kb[(size_t)n * DD + kc + k4];
        float2 lo; lo.x = f.x; lo.y = f.y;
        float2 hi; hi.x = f.z; hi.y = f.w;
        *(float2*)&Bs[(((k4 >> 1) + 0) * 512 + n) * 2] = lo;
        *(float2*)&Bs[(((k4 >> 1) + 1) * 512 + n) * 2] = hi;
      }
    }
    __syncthreads();
#pragma unroll
    for (int ks = 0; ks < 16; ks += 4) {
      v2f a = frag_a(As, 16, lane, ks);
#pragma unroll
      for (int nt = 0; nt < 4; ++nt) {
        v2f b2 = frag_b(Bs, 512, lane, ks, wv * 64 + nt * 16);
        acc[nt] = wmma4(a, b2, acc[nt]);
      }
    }
  }
  int half = lane >> 4, l16 = lane & 15;
  if (half == 0) {   // rows 0..7 are the valid slots
#pragma unroll
    for (int nt = 0; nt < 4; ++nt)
#pragma unroll
      for (int i = 0; i < 8; ++i) {
        int f = nb + wv * 64 + nt * 16 + l16;
        Dots[((size_t)b * SS + i) * FF + f] = acc[nt][i] * SCALE_C;
      }
  }
}

// softmax over slots (axis S), +EPS, atomic per-(b,s) row sums over f
__global__ void __launch_bounds__(256)
softmax_kernel(const float* __restrict__ Dots, float* __restrict__ PreNorm,
               float* __restrict__ Attn, float* __restrict__ Sums) {
  int b = blockIdx.y;
  int f = blockIdx.x * 256 + threadIdx.x;
  int lane = threadIdx.x & 31;
  float d[SS]; float mx = -3.4e38f;
#pragma unroll
  for (int s = 0; s < SS; ++s) {
    d[s] = Dots[((size_t)b * SS + s) * FF + f];
    mx = fmaxf(mx, d[s]);
  }
  float sum = 0.f;
#pragma unroll
  for (int s = 0; s < SS; ++s) { d[s] = expf(d[s] - mx); sum += d[s]; }
  float inv = 1.f / sum;
#pragma unroll
  for (int s = 0; s < SS; ++s) {
    float p = d[s] * inv;
    size_t idx = ((size_t)b * SS + s) * FF + f;
    PreNorm[idx] = p;
    float aw = p + EPS_A;
    Attn[idx] = aw;
    float r = wave_sum(aw);
    if (lane == 0) atomicAdd(&Sums[b * SS + s], r);
  }
}

__global__ void zero_sums_kernel(float* __restrict__ Sums) {
  Sums[threadIdx.x] = 0.f;
}

// updates[b,s,:] = (attn[b,s,:] @ values[b]) / Sums[b,s]   (K = F = 4096)
__global__ void __launch_bounds__(256)
updates_kernel(const float* __restrict__ Attn, const float* __restrict__ Vals,
               const float* __restrict__ Sums, float* __restrict__ Upd) {
  __shared__ float As[16 * 16];
  __shared__ float Bs[16 * 512];
  int t = threadIdx.x, lane = t & 31, wv = t >> 5;
  int b = blockIdx.x;
  const float* ab = Attn + (size_t)b * SS * FF;
  const float* vb = Vals + (size_t)b * FF * DD;
  v8f acc[4] = {};
  for (int kc = 0; kc < FF; kc += 16) {
    __syncthreads();
    {
      int r = t >> 4, k = t & 15;
      As[r * 16 + k] = (r < SS) ? ab[(size_t)r * FF + kc + k] : 0.f;
    }
    {
#pragma unroll
      for (int j = 0; j < 8; ++j) {
        int u = t + j * 256;           // 0..2047 float4 slots
        int p = u >> 8;                // k-pair 0..7
        int n2 = (u & 255) * 2;        // column pair 0..510
        float2 r0 = *(const float2*)&vb[(size_t)(kc + 2 * p) * DD + n2];
        float2 r1 = *(const float2*)&vb[(size_t)(kc + 2 * p + 1) * DD + n2];
        float4 wb; wb.x = r0.x; wb.y = r1.x; wb.z = r0.y; wb.w = r1.y;
        *(float4*)&Bs[(p * 512 + n2) * 2] = wb;
      }
    }
    __syncthreads();
#pragma unroll
    for (int ks = 0; ks < 16; ks += 4) {
      v2f a = frag_a(As, 16, lane, ks);
#pragma unroll
      for (int nt = 0; nt < 4; ++nt) {
        v2f b2 = frag_b(Bs, 512, lane, ks, wv * 64 + nt * 16);
        acc[nt] = wmma4(a, b2, acc[nt]);
      }
    }
  }
  int half = lane >> 4, l16 = lane & 15;
  if (half == 0) {
#pragma unroll
    for (int i = 0; i < 8; ++i) {
      float inv = 1.f / Sums[b * SS + i];
#pragma unroll
      for (int nt = 0; nt < 4; ++nt) {
        int n = wv * 64 + nt * 16 + l16;
        Upd[((size_t)b * SS + i) * DD + n] = acc[nt][i] * inv;
      }
    }
  }
}

// torch GRUCell gates; slots_out = (1-z)*n + z*h
__global__ void __launch_bounds__(256)
gru_kernel(const float* __restrict__ gi, const float* __restrict__ gh,
           const float* __restrict__ sn, float* __restrict__ slots) {
  int idx = blockIdx.x * 256 + threadIdx.x;   // 0 .. 256*512-1
  int m = idx >> 9, d = idx & 511;
  size_t base = (size_t)m * 3 * DD;
  float ir = gi[base + d],           hr = gh[base + d];
  float iz = gi[base + DD + d],      hz = gh[base + DD + d];
  float in_ = gi[base + 2 * DD + d], hn = gh[base + 2 * DD + d];
  float r = 1.f / (1.f + expf(-(ir + hr)));
  float z = 1.f / (1.f + expf(-(iz + hz)));
  float n = tanhf(in_ + r * hn);
  float h = sn[(size_t)m * DD + d];
  slots[(size_t)m * DD + d] = (1.f - z) * n + z * h;
}

// ---------------------------------------------------------------- host side
extern "C" void kernel_launch(void* const* d_in, const int* in_sizes, int n_in,
                              void* d_out, int out_size, void* d_ws, size_t ws_size,
                              hipStream_t stream) {
  (void)in_sizes; (void)n_in; (void)out_size; (void)ws_size;
  const float* in_slots  = (const float*)d_in[0];
  const float* features  = (const float*)d_in[1];
  const float* w_k       = (const float*)d_in[2];
  const float* w_v       = (const float*)d_in[3];
  const float* w_q       = (const float*)d_in[4];
  const float* ln_feat_w = (const float*)d_in[5];
  const float* ln_feat_b = (const float*)d_in[6];
  const float* ln_slot_w = (const float*)d_in[7];
  const float* ln_slot_b = (const float*)d_in[8];
  const float* gru_w_ih  = (const float*)d_in[9];
  const float* gru_w_hh  = (const float*)d_in[10];
  const float* gru_b_ih  = (const float*)d_in[11];
  const float* gru_b_hh  = (const float*)d_in[12];
  const float* ln_mlp_w  = (const float*)d_in[13];
  const float* ln_mlp_b  = (const float*)d_in[14];
  const float* mlp_w1    = (const float*)d_in[15];
  const float* mlp_b1    = (const float*)d_in[16];
  const float* mlp_w2    = (const float*)d_in[17];
  const float* mlp_b2    = (const float*)d_in[18];

  float* ws = (float*)d_ws;
  size_t off = 0;
  auto carve = [&](size_t n) { float* p = ws + off; off += n; return p; };
  float* mean  = carve((size_t)BB * FF);
  float* rstd  = carve((size_t)BB * FF);
  float* keys  = carve((size_t)BB * FF * DD);
  float* vals  = carve((size_t)BB * FF * DD);
  float* slots = carve((size_t)BB * SS * DD);
  float* s_n   = carve((size_t)BB * SS * DD);
  float* q     = carve((size_t)BB * SS * DD);
  float* dots  = carve((size_t)BB * SS * FF);
  float* attn  = carve((size_t)BB * SS * FF);
  float* sums  = carve(256);
  float* upd   = carve((size_t)BB * SS * DD);
  float* gi    = carve((size_t)BB * SS * 3 * DD);
  float* gh    = carve((size_t)BB * SS * 3 * DD);
  float* hln   = carve((size_t)BB * SS * DD);
  float* h1    = carve((size_t)BB * SS * HH);

  float* out_slots = (float*)d_out;
  float* out_attn  = out_slots + (size_t)BB * SS * DD;

  hipMemcpyAsync(slots, in_slots, (size_t)BB * SS * DD * sizeof(float),
                 hipMemcpyDeviceToDevice, stream);

  // feature LN stats + fused-LN dual projection (keys, values)
  row_stats_kernel<<<(BB * FF) / 8, 256, 0, stream>>>(features, mean, rstd);
  gemm_ln_dual_kernel<<<dim3((BB * FF) / 128, DD / 64), 256, 0, stream>>>(
      features, mean, rstd, ln_feat_w, ln_feat_b, w_k, w_v, keys, vals);

  for (int it = 0; it < NITER; ++it) {
    ln_rows_kernel<<<32, 256, 0, stream>>>(slots, ln_slot_w, ln_slot_b, s_n);
    gemm_rm_kernel<0><<<dim3(2, DD / 64), 256, 0, stream>>>(
        s_n, w_q, nullptr, q, nullptr, DD, DD);
    zero_sums_kernel<<<1, 256, 0, stream>>>(sums);
    dots_kernel<<<dim3(BB, FF / 512), 256, 0, stream>>>(q, keys, dots);
    softmax_kernel<<<dim3(FF / 256, BB), 256, 0, stream>>>(dots, out_attn, attn, sums);
    updates_kernel<<<BB, 256, 0, stream>>>(attn, vals, sums, upd);
    gemm_bt_kernel<<<dim3(2, (3 * DD) / 64), 256, 0, stream>>>(
        upd, gru_w_ih, gru_b_ih, gi, DD, 3 * DD);
    gemm_bt_kernel<<<dim3(2, (3 * DD) / 64), 256, 0, stream>>>(
        s_n, gru_w_hh, gru_b_hh, gh, DD, 3 * DD);
    gru_kernel<<<(BB * SS * DD) / 256, 256, 0, stream>>>(gi, gh, s_n, slots);
    ln_rows_kernel<<<32, 256, 0, stream>>>(slots, ln_mlp_w, ln_mlp_b, hln);
    gemm_rm_kernel<1><<<dim3(2, HH / 64), 256, 0, stream>>>(
        hln, mlp_w1, mlp_b1, h1, nullptr, DD, HH);
    gemm_rm_kernel<2><<<dim3(2, DD / 64), 256, 0, stream>>>(
        h1, mlp_w2, mlp_b2, slots, slots, HH, DD);
  }

  hipMemcpyAsync(out_slots, slots, (size_t)BB * SS * DD * sizeof(float),
                 hipMemcpyDeviceToDevice, stream);
}